// DepthScaleShiftInvLoss_17660905521836
// MI455X (gfx1250) — compile-verified
//
#include <hip/hip_runtime.h>

// Problem geometry (fixed by reference): B=32, H=W=512
#define NB            32
#define HW            262144                // 512*512
#define THREADS       256
#define EPT           16                    // elements per thread (4 x float4)
#define CHUNK         4096                  // THREADS * EPT
#define BLK_PER_IMG   64                    // HW / CHUNK
#define NBLK          2048                  // NB * BLK_PER_IMG
#define EPSV          1e-6f

typedef __attribute__((ext_vector_type(2))) float v2f;
typedef __attribute__((ext_vector_type(4))) float v4f;
typedef __attribute__((ext_vector_type(4))) int   v4i;
typedef __attribute__((ext_vector_type(8))) float v8f;

// ---------------------------------------------------------------------------
// Exact f32 wave32 sum using V_WMMA_F32_16X16X4_F32.
// A (16x4): lane m regs {x,0} (m<16), lane m+16 regs {x,0} -> A[m][2]=x[m+16].
// B = all ones. D[m][n] = x[m] + x[m+16] (independent of n).
// Lane layout of D: lanes 0-15 hold rows 0..7, lanes 16-31 rows 8..15,
// so sum of the 8 D regs gives a half-wave sum; one xor-16 shuffle finishes.
// ---------------------------------------------------------------------------
__device__ __forceinline__ float wave_sum_f32(float x) {
#if defined(__has_builtin) && __has_builtin(__builtin_amdgcn_wmma_f32_16x16x4_f32)
  v2f a; a[0] = x;    a[1] = 0.0f;
  v2f b; b[0] = 1.0f; b[1] = 1.0f;
  v8f c = {};
  v8f d = __builtin_amdgcn_wmma_f32_16x16x4_f32(
      /*neg_a=*/false, a, /*neg_b=*/false, b,
      /*c_mod=*/(short)0, c, /*reuse_a=*/false, /*reuse_b=*/false);
  float h = ((d[0] + d[1]) + (d[2] + d[3])) + ((d[4] + d[5]) + (d[6] + d[7]));
  h += __shfl_xor(h, 16, 32);
  return h;
#else
  float h = x;
  #pragma unroll
  for (int off = 16; off > 0; off >>= 1) h += __shfl_xor(h, off, 32);
  return h;
#endif
}

// ---------------------------------------------------------------------------
// Stage 1: per-block partial sums of pred*mask, gt*mask, mask count.
// ws1[block*4 + {0,1,2}]
// ---------------------------------------------------------------------------
__global__ __launch_bounds__(THREADS)
void dssil_stage1(const float* __restrict__ pred, const float* __restrict__ gt,
                  const int* __restrict__ mask, float* __restrict__ ws1) {
  const int t = threadIdx.x;
  const size_t base = (size_t)blockIdx.x * CHUNK;

  float sp = 0.f, sg = 0.f, sc = 0.f;
  #pragma unroll
  for (int i = 0; i < 4; ++i) {
    const size_t off = base + (size_t)i * (THREADS * 4) + (size_t)t * 4;
    const v4f p = *(const v4f*)(pred + off);
    const v4f g = *(const v4f*)(gt + off);
    const v4i m = *(const v4i*)(mask + off);
    #pragma unroll
    for (int k = 0; k < 4; ++k) {
      const float mf = m[k] ? 1.0f : 0.0f;
      sp = __builtin_fmaf(p[k], mf, sp);
      sg = __builtin_fmaf(g[k], mf, sg);
      sc += mf;
    }
  }

  const float wp = wave_sum_f32(sp);
  const float wg = wave_sum_f32(sg);
  const float wc = wave_sum_f32(sc);

  __shared__ float wsum[8][4];
  const int wave = t >> 5, lane = t & 31;
  if (lane == 0) { wsum[wave][0] = wp; wsum[wave][1] = wg; wsum[wave][2] = wc; }
  __syncthreads();
  if (t == 0) {
    float a = 0.f, b = 0.f, c = 0.f;
    #pragma unroll
    for (int w = 0; w < 8; ++w) { a += wsum[w][0]; b += wsum[w][1]; c += wsum[w][2]; }
    float* r = ws1 + (size_t)blockIdx.x * 4;
    r[0] = a; r[1] = b; r[2] = c;
  }
}

// ---------------------------------------------------------------------------
// Stage 2: recover shifts (deterministic fixed-order reduce of 64 partials),
// then per-block partial sums of |x - shift| * mask.
// ws2[block*2 + {0,1}], ws3[b*4 + {shift_p, shift_g, count}]
// ---------------------------------------------------------------------------
__global__ __launch_bounds__(THREADS)
void dssil_stage2(const float* __restrict__ pred, const float* __restrict__ gt,
                  const int* __restrict__ mask, const float* __restrict__ ws1,
                  float* __restrict__ ws2, float* __restrict__ ws3) {
  __shared__ float sh[3][64];
  __shared__ float stats[3];
  const int t = threadIdx.x;
  const int b = blockIdx.x >> 6;  // / BLK_PER_IMG

  if (t < 64) {
    const float* r = ws1 + (size_t)(b * BLK_PER_IMG + t) * 4;
    sh[0][t] = r[0]; sh[1][t] = r[1]; sh[2][t] = r[2];
  }
  __syncthreads();
  #pragma unroll
  for (int s = 32; s > 0; s >>= 1) {
    if (t < s) { sh[0][t] += sh[0][t+s]; sh[1][t] += sh[1][t+s]; sh[2][t] += sh[2][t+s]; }
    __syncthreads();
  }
  if (t == 0) {
    const float cnt = fmaxf(sh[2][0], 1.0f);
    stats[0] = sh[0][0] / cnt; stats[1] = sh[1][0] / cnt; stats[2] = cnt;
  }
  __syncthreads();
  const float shift_p = stats[0], shift_g = stats[1];

  const size_t base = (size_t)blockIdx.x * CHUNK;
  float ap = 0.f, ag = 0.f;
  #pragma unroll
  for (int i = 0; i < 4; ++i) {
    const size_t off = base + (size_t)i * (THREADS * 4) + (size_t)t * 4;
    const v4f p = *(const v4f*)(pred + off);
    const v4f g = *(const v4f*)(gt + off);
    const v4i m = *(const v4i*)(mask + off);
    #pragma unroll
    for (int k = 0; k < 4; ++k) {
      const float mf = m[k] ? 1.0f : 0.0f;
      ap = __builtin_fmaf(fabsf(p[k] - shift_p), mf, ap);
      ag = __builtin_fmaf(fabsf(g[k] - shift_g), mf, ag);
    }
  }

  const float wap = wave_sum_f32(ap);
  const float wag = wave_sum_f32(ag);

  __shared__ float wsum[8][2];
  const int wave = t >> 5, lane = t & 31;
  if (lane == 0) { wsum[wave][0] = wap; wsum[wave][1] = wag; }
  __syncthreads();
  if (t == 0) {
    float a = 0.f, c = 0.f;
    #pragma unroll
    for (int w = 0; w < 8; ++w) { a += wsum[w][0]; c += wsum[w][1]; }
    float* r = ws2 + (size_t)blockIdx.x * 2;
    r[0] = a; r[1] = c;
    // 64 blocks of the same image write identical values: benign.
    float* s3 = ws3 + (size_t)b * 4;
    s3[0] = shift_p; s3[1] = shift_g; s3[2] = stats[2];
  }
}

// ---------------------------------------------------------------------------
// Stage 3: recover scales (same fixed-order reduce), then elementwise output.
// Output stored non-temporally (write-once, keep inputs resident in L2).
// ---------------------------------------------------------------------------
__global__ __launch_bounds__(THREADS)
void dssil_stage3(const float* __restrict__ pred, const float* __restrict__ gt,
                  const int* __restrict__ mask, const float* __restrict__ ws2,
                  const float* __restrict__ ws3, float* __restrict__ out) {
  __shared__ float sh[2][64];
  __shared__ float stats[4];
  const int t = threadIdx.x;
  const int b = blockIdx.x >> 6;

  if (t < 64) {
    const float* r = ws2 + (size_t)(b * BLK_PER_IMG + t) * 2;
    sh[0][t] = r[0]; sh[1][t] = r[1];
  }
  __syncthreads();
  #pragma unroll
  for (int s = 32; s > 0; s >>= 1) {
    if (t < s) { sh[0][t] += sh[0][t+s]; sh[1][t] += sh[1][t+s]; }
    __syncthreads();
  }
  if (t == 0) {
    const float* s3 = ws3 + (size_t)b * 4;
    const float cnt = s3[2];
    stats[0] = s3[0];                              // shift_p
    stats[1] = s3[1];                              // shift_g
    stats[2] = 1.0f / fmaxf(sh[0][0] / cnt, EPSV); // 1/scale_p
    stats[3] = 1.0f / fmaxf(sh[1][0] / cnt, EPSV); // 1/scale_g
  }
  __syncthreads();
  const float shift_p = stats[0], shift_g = stats[1];
  const float isp = stats[2], isg = stats[3];

  const size_t base = (size_t)blockIdx.x * CHUNK;
  #pragma unroll
  for (int i = 0; i < 4; ++i) {
    const size_t off = base + (size_t)i * (THREADS * 4) + (size_t)t * 4;
    const v4f p = *(const v4f*)(pred + off);
    const v4f g = *(const v4f*)(gt + off);
    const v4i m = *(const v4i*)(mask + off);
    v4f o;
    #pragma unroll
    for (int k = 0; k < 4; ++k) {
      const float mf = m[k] ? 1.0f : 0.0f;
      o[k] = fabsf((p[k] - shift_p) * isp - (g[k] - shift_g) * isg) * mf;
    }
    __builtin_nontemporal_store(o, (v4f*)(out + off));
  }
}

// ---------------------------------------------------------------------------
extern "C" void kernel_launch(void* const* d_in, const int* in_sizes, int n_in,
                              void* d_out, int out_size, void* d_ws, size_t ws_size,
                              hipStream_t stream) {
  (void)in_sizes; (void)n_in; (void)out_size; (void)ws_size;
  const float* pred = (const float*)d_in[0];
  const float* gt   = (const float*)d_in[1];
  const int*   mask = (const int*)d_in[2];
  float* out = (float*)d_out;

  float* ws1 = (float*)d_ws;            // NBLK*4 partials (sum_p, sum_g, cnt)
  float* ws2 = ws1 + (size_t)NBLK * 4;  // NBLK*2 partials (abs_p, abs_g)
  float* ws3 = ws2 + (size_t)NBLK * 2;  // NB*4 stats (shift_p, shift_g, count)

  dim3 grid(NBLK), block(THREADS);
  dssil_stage1<<<grid, block, 0, stream>>>(pred, gt, mask, ws1);
  dssil_stage2<<<grid, block, 0, stream>>>(pred, gt, mask, ws1, ws2, ws3);
  dssil_stage3<<<grid, block, 0, stream>>>(pred, gt, mask, ws2, ws3, out);
}